// GCNLayer_22565758173846
// MI455X (gfx1250) — compile-verified
//
#include <hip/hip_runtime.h>
#include <hip/hip_bf16.h>

typedef __attribute__((ext_vector_type(2))) float v2f;
typedef __attribute__((ext_vector_type(8))) float v8f;

#define F 128            // IN_FEATS == OUT_FEATS
#define LDSW 132         // padded LDS row stride (dwords) -> conflict-free b64 reads
#define EDGES_PER_WAVE 4 // consecutive edges per wave in the scatter kernel

// ---------------------------------------------------------------------------
// Kernel 1: zero the aggregation buffer (d_ws is poisoned by the harness).
// ---------------------------------------------------------------------------
__global__ void gcn_zero_kernel(float4* __restrict__ p, int n4) {
    int i = blockIdx.x * blockDim.x + threadIdx.x;
    if (i < n4) p[i] = make_float4(0.f, 0.f, 0.f, 0.f);
}

// ---------------------------------------------------------------------------
// Kernel 2: edge scatter. One wave32 per run of EDGES_PER_WAVE consecutive
// edges; each lane handles one float4 (128 floats / 32 lanes). The out_norm
// divide is folded into the message so h is never materialized. feat (51MB)
// and agg (51MB) both fit in the 192MB L2, so the random gather and the f32
// atomics resolve in-cache. The next edge's source row is prefetched
// (global_prefetch_b8) while the current edge's atomics are in flight.
// ---------------------------------------------------------------------------
__global__ void gcn_edge_kernel(const float* __restrict__ feat,
                                const float* __restrict__ out_norm,
                                const int* __restrict__ src,
                                const int* __restrict__ dst,
                                float* __restrict__ agg,
                                int n_edges) {
    int tid  = blockIdx.x * blockDim.x + threadIdx.x;
    int wid  = tid >> 5;          // wave32 id
    int lane = tid & 31;

    int e0 = wid * EDGES_PER_WAVE;
    if (e0 >= n_edges) return;
    int e1 = min(e0 + EDGES_PER_WAVE, n_edges);

    int s = src[e0];
    for (int e = e0; e < e1; ++e) {
        int d = dst[e];
        float inv = 1.0f / out_norm[s];

        const float4* frow = (const float4*)(feat + (size_t)s * F);
        float4 v = frow[lane];

        // Prefetch the next edge's source row (one byte per lane at lane*16
        // spans all cachelines of the 512B row).
        if (e + 1 < e1) {
            s = src[e + 1];
            __builtin_prefetch((const char*)(feat + (size_t)s * F) + lane * 16, 0, 3);
        }

        v.x *= inv; v.y *= inv; v.z *= inv; v.w *= inv;

        float* arow = agg + (size_t)d * F + lane * 4;
        atomicAdd(arow + 0, v.x);
        atomicAdd(arow + 1, v.y);
        atomicAdd(arow + 2, v.z);
        atomicAdd(arow + 3, v.w);
    }
}

// ---------------------------------------------------------------------------
// Kernel 3: out = (agg / in_norm) @ W^T + b  via V_WMMA_F32_16X16X4_F32.
// One wave per 32-row super-tile (2 x 16-row M tiles). The 128 output
// columns are processed in TWO sequential halves of 64 columns so only
// 8 v8f accumulators (64 VGPRs) are live at once -> no spills, while each
// B fragment loaded from LDS still feeds TWO wmma ops (one per M tile):
// half the DS traffic per matrix op vs. one-tile-per-wave.
//
// f32 WMMA fragment layout (ISA 7.12.2):
//   A (16x4):  lanes 0-15 hold M=0..15 {K0,K1}; lanes 16-31 hold {K2,K3}.
//   B (4x16):  lanes 0-15 hold N=0..15 {K0,K1}; lanes 16-31 hold {K2,K3}.
//   C/D:       VGPR r: lanes 0-15 = D[M=r][N=lane], lanes 16-31 = D[M=r+8].
// W staged in LDS row-major, row stride 132 dwords: bank = (4n + k) % 64,
// so the wave's b64 B-load hits 64 distinct banks.
// ---------------------------------------------------------------------------
__global__ void __launch_bounds__(256, 1)
gcn_wmma_gemm_kernel(const float* __restrict__ agg,
                     const float* __restrict__ in_norm,
                     const float* __restrict__ W,
                     const float* __restrict__ bias,
                     float* __restrict__ out,
                     int n_nodes) {
    extern __shared__ float lds_w[];   // F * LDSW floats = 67584 bytes

    // Cooperative stage of W[128][128] into padded LDS.
    for (int i = threadIdx.x; i < (F * F) / 4; i += blockDim.x) {
        int row = (i * 4) / F;
        int col = (i * 4) % F;
        float4 v = ((const float4*)W)[i];
        float* p = &lds_w[row * LDSW + col];
        p[0] = v.x; p[1] = v.y; p[2] = v.z; p[3] = v.w;
    }
    __syncthreads();

    int wave  = threadIdx.x >> 5;
    int lane  = threadIdx.x & 31;
    int stile = blockIdx.x * 8 + wave;         // 8 waves per 256-thread block
    int row0  = stile * 32;                    // 32 rows per wave
    if (row0 >= n_nodes) return;               // wave-uniform -> EXEC all-1s
    bool two = (row0 + 16) < n_nodes;          // second M tile valid (uniform)

    int nlo   = lane & 15;
    int khalf = (lane >> 4) * 2;               // 0 for lanes 0-15, 2 for 16-31
    int m0    = row0 + nlo;
    int m1    = row0 + 16 + nlo;
    float inv0 = 1.0f / in_norm[m0];
    float inv1 = two ? (1.0f / in_norm[m1]) : 0.0f;

    const float* arow0 = agg + (size_t)m0 * F + khalf;
    const float* arow1 = agg + (size_t)(two ? m1 : m0) * F + khalf;
    int moff = (lane >> 4) * 8;                // D row offset for this lane half

    for (int nh = 0; nh < 2; ++nh) {           // two 64-column halves
        int ncol0 = nh * 64;

        v8f acc0[4] = {};                      // rows [row0,      row0 + 16)
        v8f acc1[4] = {};                      // rows [row0 + 16, row0 + 32)

        for (int k0 = 0; k0 < F; k0 += 4) {
            v2f a0 = *(const v2f*)(arow0 + k0);
            v2f a1 = *(const v2f*)(arow1 + k0);
            a0 *= inv0;
            a1 *= inv1;
#pragma unroll
            for (int nb = 0; nb < 4; ++nb) {
                int n = ncol0 + nb * 16 + nlo;
                v2f bf = *(const v2f*)(&lds_w[n * LDSW + k0 + khalf]);
                acc0[nb] = __builtin_amdgcn_wmma_f32_16x16x4_f32(
                    false, a0, false, bf, (short)0, acc0[nb], false, false);
                acc1[nb] = __builtin_amdgcn_wmma_f32_16x16x4_f32(
                    false, a1, false, bf, (short)0, acc1[nb], false, false);
            }
        }

        // Epilogue for this half: bias add + store.
        // D VGPR r -> row (base + r + 8*(lane>=16)), col n.
#pragma unroll
        for (int nb = 0; nb < 4; ++nb) {
            int n = ncol0 + nb * 16 + nlo;
            float bb = bias[n];
            float* op0 = out + (size_t)(row0 + moff) * F + n;
#pragma unroll
            for (int r = 0; r < 8; ++r) {
                op0[(size_t)r * F] = acc0[nb][r] + bb;
            }
            if (two) {
                float* op1 = out + (size_t)(row0 + 16 + moff) * F + n;
#pragma unroll
                for (int r = 0; r < 8; ++r) {
                    op1[(size_t)r * F] = acc1[nb][r] + bb;
                }
            }
        }
    }
}

// ---------------------------------------------------------------------------
// Host-side launcher.
// Inputs (setup_inputs order): feat[N*128], in_norm[N], out_norm[N],
//                              src[E], dst[E], W[128*128], b[128]
// ---------------------------------------------------------------------------
extern "C" void kernel_launch(void* const* d_in, const int* in_sizes, int n_in,
                              void* d_out, int out_size, void* d_ws, size_t ws_size,
                              hipStream_t stream) {
    const float* feat     = (const float*)d_in[0];
    const float* in_norm  = (const float*)d_in[1];
    const float* out_norm = (const float*)d_in[2];
    const int*   src      = (const int*)d_in[3];
    const int*   dst      = (const int*)d_in[4];
    const float* W        = (const float*)d_in[5];
    const float* bias     = (const float*)d_in[6];
    float*       out      = (float*)d_out;

    int n_nodes = in_sizes[1];     // 100000
    int n_edges = in_sizes[3];     // 1600000

    float* agg = (float*)d_ws;     // n_nodes * 128 floats = 51.2 MB scratch

    // 1) agg = 0
    {
        int n4 = (n_nodes * F) / 4;
        int blocks = (n4 + 255) / 256;
        gcn_zero_kernel<<<blocks, 256, 0, stream>>>((float4*)agg, n4);
    }

    // 2) scatter-add edges (L2-resident atomics, prefetch-pipelined gather)
    {
        long long nwaves = ((long long)n_edges + EDGES_PER_WAVE - 1) / EDGES_PER_WAVE;
        long long threads = nwaves * 32;
        int blocks = (int)((threads + 255) / 256);
        gcn_edge_kernel<<<blocks, 256, 0, stream>>>(feat, out_norm, src, dst,
                                                    agg, n_edges);
    }

    // 3) WMMA GEMM epilogue (2 M-tiles per wave, 2 sequential N-halves)
    {
        int stiles = (n_nodes + 31) / 32;      // 3125, exact
        int blocks = (stiles + 7) / 8;
        size_t shmem = (size_t)F * LDSW * sizeof(float);   // 67584 B
        gcn_wmma_gemm_kernel<<<blocks, 256, shmem, stream>>>(agg, in_norm, W,
                                                             bias, out, n_nodes);
    }
}